// GCN_7026566496344
// MI455X (gfx1250) — compile-verified
//
#include <hip/hip_runtime.h>

// GCN 3-stage pipeline for gfx1250 (MI455X), fp32 end-to-end.
// Dense 64-wide GEMMs use V_WMMA_F32_16X16X4_F32 (compile-time specialized tiles);
// edge scatter uses HW fp32 global atomics. Working set (~60 MB) is L2-resident.

#define LEAKY_SLOPE 0.01f
#define BN_EPS 1e-5f

typedef float v2f __attribute__((ext_vector_type(2)));
typedef float v8f __attribute__((ext_vector_type(8)));

__device__ __forceinline__ float leakyf(float v) { return v >= 0.0f ? v : LEAKY_SLOPE * v; }

__device__ __forceinline__ void atomicAddF32(float* p, float v) {
  // HW global_atomic_add_f32
  unsafeAtomicAdd(p, v);
}

// ---------------- degree / normalization ----------------
__global__ void k_deg_init(float* deg, int n) {
  int i = blockIdx.x * blockDim.x + threadIdx.x;
  if (i < n) deg[i] = 2.0f;  // improved=True self-loop weight
}

__global__ void k_deg_count(const int* __restrict__ dst, float* __restrict__ deg, int E) {
  int e = blockIdx.x * blockDim.x + threadIdx.x;
  if (e < E) atomicAddF32(&deg[dst[e]], 1.0f);
}

__global__ void k_deg_fin(float* deg, int n) {
  int i = blockIdx.x * blockDim.x + threadIdx.x;
  if (i < n) deg[i] = rsqrtf(deg[i]);  // dinv
}

// ---------------- stage-0 lift: [N,3] @ [3,64] (K too small for WMMA) ----------------
__global__ void k_lift3(const float* __restrict__ x, const float* __restrict__ w,
                        float* __restrict__ h, int n) {
  int idx = blockIdx.x * blockDim.x + threadIdx.x;
  if (idx >= n * 64) return;
  int node = idx >> 6, j = idx & 63;
  const float* xr = x + node * 3;
  h[idx] = xr[0] * w[j] + xr[1] * w[64 + j] + xr[2] * w[128 + j];
}

// ---------------- WMMA fp32 GEMM: C[n,NCOLS] = A[n,64] x B, optional bias+leaky ----
// TRANSB==0: B is [64, NCOLS] row-major  (B[k][j] = B[k*NCOLS + j])
// TRANSB==1: B is [NCOLS, 64] and we use B^T (B[k][j] = B[j*64 + k])  -> h @ lw.T
// Fully compile-time specialized: no runtime guards in the K-loop.
template <int NCOLS, int TRANSB, int HASBIAS, int DOLEAKY>
__global__ void k_wmma_gemm(const float* __restrict__ A, const float* __restrict__ B,
                            const float* __restrict__ bias, float* __restrict__ C,
                            int nrows) {
  constexpr int LDB = TRANSB ? 64 : NCOLS;
  constexpr int NTILES = (NCOLS + 15) / 16;
  constexpr bool PARTIAL = (NCOLS % 16) != 0;

  int wave = (int)((blockIdx.x * blockDim.x + threadIdx.x) >> 5);
  int lane = threadIdx.x & 31;
  int mtile = wave / NTILES;
  int ntile = wave % NTILES;
  if (mtile * 16 >= nrows) return;  // uniform per wave -> EXEC stays all-ones

  int m  = mtile * 16 + (lane & 15);
  int nn = ntile * 16 + (lane & 15);
  // clamped column for loads (always in-bounds); zero-select handles padding
  int nnl = PARTIAL ? (nn < NCOLS ? nn : NCOLS - 1) : nn;
  bool nok = PARTIAL ? (nn < NCOLS) : true;
  int ko = (lane < 16) ? 0 : 2;  // K-halves per ISA 16x4 A / 4x16 B layouts

  const float* arow = A + (long long)m * 64;
  v8f acc = {};
#pragma unroll
  for (int kb = 0; kb < 64; kb += 4) {
    int k0 = kb + ko;
    v2f a = *(const v2f*)(arow + k0);  // 8B aligned (k0 even)
    float bx, by;
    if (TRANSB) {
      bx = B[nnl * LDB + k0];
      by = B[nnl * LDB + k0 + 1];
    } else {
      bx = B[k0 * LDB + nnl];
      by = B[(k0 + 1) * LDB + nnl];
    }
    v2f b;
    b.x = nok ? bx : 0.0f;   // v_cndmask, no EXEC branching
    b.y = nok ? by : 0.0f;
    acc = __builtin_amdgcn_wmma_f32_16x16x4_f32(
        /*neg_a=*/false, a, /*neg_b=*/false, b,
        /*c_mod=*/(short)0, acc, /*reuse_a=*/false, /*reuse_b=*/false);
  }

  float bv = HASBIAS ? bias[nnl] : 0.0f;
#pragma unroll
  for (int r = 0; r < 8; ++r) {
    int row = mtile * 16 + ((lane < 16) ? r : (r + 8));
    float v = acc[r] + bv;
    if (DOLEAKY) v = leakyf(v);
    if (nok) C[(long long)row * NCOLS + nn] = v;
  }
}

// ---------------- self-loop + bias init of aggregation buffer ----------------
template <int C>
__global__ void k_self_init(const float* __restrict__ h, const float* __restrict__ dinv,
                            const float* __restrict__ bias, float* __restrict__ agg, int n) {
  long long idx = (long long)blockIdx.x * blockDim.x + threadIdx.x;
  if (idx >= (long long)n * C) return;
  int node = (int)(idx / C);
  int c    = (int)(idx % C);
  float di = dinv[node];
  agg[idx] = 2.0f * di * di * h[idx] + bias[c];
}

// ---------------- edge scatter-add: agg[dst] += h[src] * dinv[src]*dinv[dst] --------
template <int C, int CPAD>
__global__ void k_edge(const int* __restrict__ src, const int* __restrict__ dst,
                       const float* __restrict__ dinv, const float* __restrict__ h,
                       float* __restrict__ agg, int E) {
  long long idx = (long long)blockIdx.x * blockDim.x + threadIdx.x;
  int e = (int)(idx / CPAD);
  int c = (int)(idx % CPAD);
  if (e >= E || c >= C) return;
  int s = src[e], d = dst[e];
  float w = dinv[s] * dinv[d];
  atomicAddF32(&agg[(long long)d * C + c], h[(long long)s * C + c] * w);
}

// ---------------- BatchNorm: stats (one block per channel), finalize, apply+pool ----
// stats layout: [0..63]=sum  [64..127]=sumsq  [128..191]=scale  [192..255]=shift
__global__ void k_bn_stats(const float* __restrict__ x, float* __restrict__ stats,
                           int n, int C) {
  int c = blockIdx.x;  // one block per channel
  float s = 0.0f, s2 = 0.0f;
  for (int i = threadIdx.x; i < n; i += 256) {
    float v = x[(long long)i * C + c];
    s += v; s2 += v * v;
  }
  __shared__ float sh[256], sh2[256];
  sh[threadIdx.x] = s; sh2[threadIdx.x] = s2;
  __syncthreads();
  for (int off = 128; off > 0; off >>= 1) {
    if ((int)threadIdx.x < off) {
      sh[threadIdx.x]  += sh[threadIdx.x + off];
      sh2[threadIdx.x] += sh2[threadIdx.x + off];
    }
    __syncthreads();
  }
  if (threadIdx.x == 0) { stats[c] = sh[0]; stats[64 + c] = sh2[0]; }
}

__global__ void k_bn_final(const float* __restrict__ gamma, const float* __restrict__ beta,
                           float* __restrict__ stats, int n, int C) {
  int c = threadIdx.x;
  if (c >= C) return;
  float inv_n = 1.0f / (float)n;
  float mean = stats[c] * inv_n;
  float var  = stats[64 + c] * inv_n - mean * mean;  // biased variance
  float sc   = gamma[c] * rsqrtf(var + BN_EPS);
  stats[128 + c] = sc;
  stats[192 + c] = beta[c] - mean * sc;
}

// BN-apply + MaxPool1d(window 2p+1, stride 1, pad p along channels) + LeakyReLU
template <int C>
__global__ void k_bn_pool(const float* __restrict__ x, const float* __restrict__ stats,
                          float* __restrict__ y, int n, int p) {
  long long idx = (long long)blockIdx.x * blockDim.x + threadIdx.x;
  if (idx >= (long long)n * C) return;
  int node = (int)(idx / C);
  int c    = (int)(idx % C);
  int lo = c - p; if (lo < 0) lo = 0;
  int hi = c + p; if (hi > C - 1) hi = C - 1;
  const float* xr = x + (long long)node * C;
  float m = -3.402823e38f;
  for (int w = lo; w <= hi; ++w)
    m = fmaxf(m, xr[w] * stats[128 + w] + stats[192 + w]);
  y[idx] = leakyf(m);
}

// ---------------- final 3x3 linear (y = x @ lw.T + lb, leaky) ----------------
__global__ void k_lin3(const float* __restrict__ x, const float* __restrict__ lw,
                       const float* __restrict__ lb, float* __restrict__ y, int n) {
  int node = blockIdx.x * blockDim.x + threadIdx.x;
  if (node >= n) return;
  float a0 = x[node * 3 + 0], a1 = x[node * 3 + 1], a2 = x[node * 3 + 2];
#pragma unroll
  for (int j = 0; j < 3; ++j) {
    float v = a0 * lw[j * 3 + 0] + a1 * lw[j * 3 + 1] + a2 * lw[j * 3 + 2] + lb[j];
    y[node * 3 + j] = leakyf(v);
  }
}

static inline int nblk(long long total, int t) { return (int)((total + t - 1) / t); }

extern "C" void kernel_launch(void* const* d_in, const int* in_sizes, int n_in,
                              void* d_out, int out_size, void* d_ws, size_t ws_size,
                              hipStream_t stream) {
  const float* x  = (const float*)d_in[0];
  const int*   ei = (const int*)d_in[1];
  const float *w0 = (const float*)d_in[2],  *b0 = (const float*)d_in[3];
  const float *g0 = (const float*)d_in[4],  *bb0 = (const float*)d_in[5];
  const float *lw0 = (const float*)d_in[6], *lb0 = (const float*)d_in[7];
  const float *w1 = (const float*)d_in[8],  *b1 = (const float*)d_in[9];
  const float *g1 = (const float*)d_in[10], *bb1 = (const float*)d_in[11];
  const float *lw1 = (const float*)d_in[12], *lb1 = (const float*)d_in[13];
  const float *w2 = (const float*)d_in[14], *b2 = (const float*)d_in[15];
  const float *g2 = (const float*)d_in[16], *bb2 = (const float*)d_in[17];
  const float *lw2 = (const float*)d_in[18], *lb2 = (const float*)d_in[19];

  const int N = in_sizes[0] / 3;   // 80000 (multiple of 16)
  const int E = in_sizes[1] / 2;   // 1280000
  const int* src = ei;
  const int* dst = ei + E;

  // workspace layout (floats): dinv[N] | bufH[N*64] | bufAgg[N*64] | bufOut[N*64] | stats[256]
  float* ws     = (float*)d_ws;
  float* dinv   = ws;
  float* bufH   = dinv + N;
  float* bufAgg = bufH + (long long)N * 64;
  float* bufOut = bufAgg + (long long)N * 64;
  float* stats  = bufOut + (long long)N * 64;

  const int T = 256;
  const long long gemmThr64 = (long long)(N / 16) * 4 * 32;  // 16x16 tiles, 4 n-tiles
  const long long gemmThr3  = (long long)(N / 16) * 1 * 32;

  // ---- degrees (shared by all stages) ----
  k_deg_init<<<nblk(N, T), T, 0, stream>>>(dinv, N);
  k_deg_count<<<nblk(E, T), T, 0, stream>>>(dst, dinv, E);
  k_deg_fin<<<nblk(N, T), T, 0, stream>>>(dinv, N);

  // ---- stage 0: 3 -> 64, pool k=3 p=1 ----
  k_lift3<<<nblk((long long)N * 64, T), T, 0, stream>>>(x, w0, bufH, N);
  k_self_init<64><<<nblk((long long)N * 64, T), T, 0, stream>>>(bufH, dinv, b0, bufAgg, N);
  k_edge<64, 64><<<nblk((long long)E * 64, T), T, 0, stream>>>(src, dst, dinv, bufH, bufAgg, E);
  k_bn_stats<<<64, 256, 0, stream>>>(bufAgg, stats, N, 64);
  k_bn_final<<<1, 64, 0, stream>>>(g0, bb0, stats, N, 64);
  k_bn_pool<64><<<nblk((long long)N * 64, T), T, 0, stream>>>(bufAgg, stats, bufH, N, 1);
  k_wmma_gemm<64, 1, 1, 1><<<nblk(gemmThr64, T), T, 0, stream>>>(bufH, lw0, lb0, bufOut, N);

  // ---- stage 1: 64 -> 64, pool k=5 p=2 ----
  k_wmma_gemm<64, 0, 0, 0><<<nblk(gemmThr64, T), T, 0, stream>>>(bufOut, w1, nullptr, bufH, N);
  k_self_init<64><<<nblk((long long)N * 64, T), T, 0, stream>>>(bufH, dinv, b1, bufAgg, N);
  k_edge<64, 64><<<nblk((long long)E * 64, T), T, 0, stream>>>(src, dst, dinv, bufH, bufAgg, E);
  k_bn_stats<<<64, 256, 0, stream>>>(bufAgg, stats, N, 64);
  k_bn_final<<<1, 64, 0, stream>>>(g1, bb1, stats, N, 64);
  k_bn_pool<64><<<nblk((long long)N * 64, T), T, 0, stream>>>(bufAgg, stats, bufH, N, 2);
  k_wmma_gemm<64, 1, 1, 1><<<nblk(gemmThr64, T), T, 0, stream>>>(bufH, lw1, lb1, bufOut, N);

  // ---- stage 2: 64 -> 3, pool k=3 p=1 ----
  k_wmma_gemm<3, 0, 0, 0><<<nblk(gemmThr3, T), T, 0, stream>>>(bufOut, w2, nullptr, bufH, N);
  k_self_init<3><<<nblk((long long)N * 3, T), T, 0, stream>>>(bufH, dinv, b2, bufAgg, N);
  k_edge<3, 4><<<nblk((long long)E * 4, T), T, 0, stream>>>(src, dst, dinv, bufH, bufAgg, E);
  k_bn_stats<<<3, 256, 0, stream>>>(bufAgg, stats, N, 3);
  k_bn_final<<<1, 64, 0, stream>>>(g2, bb2, stats, N, 3);
  k_bn_pool<3><<<nblk((long long)N * 3, T), T, 0, stream>>>(bufAgg, stats, bufH, N, 1);
  k_lin3<<<nblk(N, T), T, 0, stream>>>(bufH, lw2, lb2, (float*)d_out, N);
}